// ScaledDotProductAttention_38010460569701
// MI455X (gfx1250) — compile-verified
//
#include <hip/hip_runtime.h>
#include <math.h>
#include <stdint.h>

typedef __attribute__((ext_vector_type(2))) float v2f;
typedef __attribute__((ext_vector_type(8))) float v8f;

// Problem constants (match reference)
#define SEQ    2048
#define DHEAD  64
#define BHEADS 32          // B*H = 2*16

#define QTILE_BLOCK 128    // query rows per block (8 waves x 16 rows)
#define NWAVES      8
#define KTILE       64     // keys staged per LDS tile
#define KPAD        68     // K/V LDS row stride (floats): 272B rows -> 16B aligned, conflict-free
#define PPAD        66     // per-wave P scratch row stride (floats): even -> 8B aligned b64 loads

// Butterfly max across the 16-lane half (lanes 0-15 / 16-31 reduce independently).
__device__ __forceinline__ float half16_max(float x) {
#if __has_builtin(__builtin_amdgcn_permlane16)
    unsigned u = __float_as_uint(x);
    float y;
    y = __uint_as_float(__builtin_amdgcn_permlane16(u, u, 0x67452301u, 0xEFCDAB89u, false, false)); // lane^1
    x = fmaxf(x, y); u = __float_as_uint(x);
    y = __uint_as_float(__builtin_amdgcn_permlane16(u, u, 0x54761032u, 0xDCFE98BAu, false, false)); // lane^2
    x = fmaxf(x, y); u = __float_as_uint(x);
    y = __uint_as_float(__builtin_amdgcn_permlane16(u, u, 0x32107654u, 0xBA98FEDCu, false, false)); // lane^4
    x = fmaxf(x, y); u = __float_as_uint(x);
    y = __uint_as_float(__builtin_amdgcn_permlane16(u, u, 0xFEDCBA98u, 0x76543210u, false, false)); // lane^8
    return fmaxf(x, y);
#else
    float y;
    y = __int_as_float(__builtin_amdgcn_ds_swizzle(__float_as_int(x), 0x041F)); x = fmaxf(x, y);
    y = __int_as_float(__builtin_amdgcn_ds_swizzle(__float_as_int(x), 0x081F)); x = fmaxf(x, y);
    y = __int_as_float(__builtin_amdgcn_ds_swizzle(__float_as_int(x), 0x101F)); x = fmaxf(x, y);
    y = __int_as_float(__builtin_amdgcn_ds_swizzle(__float_as_int(x), 0x201F)); x = fmaxf(x, y);
    return x;
#endif
}

__device__ __forceinline__ float exp2_(float x) {
#if __has_builtin(__builtin_amdgcn_exp2f)
    return __builtin_amdgcn_exp2f(x);   // v_exp_f32 is native exp2
#else
    return exp2f(x);
#endif
}

__global__ __launch_bounds__(256, 2)
void fa_fwd_f32_wmma(const float* __restrict__ Q,
                     const float* __restrict__ K,
                     const float* __restrict__ V,
                     float* __restrict__ O)
{
    __shared__ float Ks[2][KTILE * KPAD];   // double-buffered K tile
    __shared__ float Vs[2][KTILE * KPAD];   // double-buffered V tile
    __shared__ float Ps[NWAVES * 16 * PPAD];

    const int tid  = threadIdx.x;
    const int lane = tid & 31;
    const int wave = tid >> 5;
    const int half = lane >> 4;   // 0: lanes 0-15, 1: lanes 16-31
    const int l16  = lane & 15;

    const int bh = blockIdx.y;
    const int qb = blockIdx.x * QTILE_BLOCK;
    const int q0 = qb + wave * 16;                 // this wave's first query row
    // q0 is identical in every lane of the wave; readfirstlane pins it in an SGPR so
    // all tile guards lower to scalar branches (EXEC provably all-ones at every WMMA).
    const int q15 = __builtin_amdgcn_readfirstlane(q0 + 15);
    const size_t base = (size_t)bh * SEQ * DHEAD;

    // ---- Q fragments pre-scaled by log2(e)/sqrt(D): softmax done in exp2 domain.
    // A layout (32-bit 16x4): lane<16 -> M=lane, K={0,1}; lane>=16 -> M=lane-16, K={2,3}.
    const float softmax_scale = 0.125f * 1.44269504088896340736f;
    v2f qf[16];
    {
        const float* qrowp = Q + base + (size_t)(q0 + l16) * DHEAD;
        #pragma unroll
        for (int kb = 0; kb < 16; ++kb) {
            const int k = 4 * kb + 2 * half;
            qf[kb].x = qrowp[k]     * softmax_scale;
            qf[kb].y = qrowp[k + 1] * softmax_scale;
        }
    }

    v8f oacc[4];                           // four 16x16 d-tiles, row M = r + 8*half
    #pragma unroll
    for (int t = 0; t < 4; ++t) oacc[t] = (v8f)(0.0f);
    v8f lacc = (v8f)(0.0f);                // row-sums via P x ones
    float mrun[8];
    #pragma unroll
    for (int r = 0; r < 8; ++r) mrun[r] = -__builtin_inff();

    float* const pwave = &Ps[wave * 16 * PPAD];
    v2f ones; ones.x = 1.0f; ones.y = 1.0f;

    // ---- async stage of a 64x64 K/V tile into LDS buffer `buf` (CDNA5 async-to-LDS path)
    auto stage_tile = [&](int kts, int buf) {
        const unsigned long long kg = (unsigned long long)(uintptr_t)(K + base + (size_t)kts * KTILE * DHEAD);
        const unsigned long long vg = (unsigned long long)(uintptr_t)(V + base + (size_t)kts * KTILE * DHEAD);
        #pragma unroll
        for (int i = 0; i < 4; ++i) {
            const int slot = tid + i * 256;          // 0..1023 float4 slots
            const int row  = slot >> 4;
            const int c4   = slot & 15;
            const unsigned goff = (unsigned)((row * DHEAD + c4 * 4) * 4);   // bytes
            const unsigned ldsK = (unsigned)(uintptr_t)&Ks[buf][row * KPAD + c4 * 4];
            const unsigned ldsV = (unsigned)(uintptr_t)&Vs[buf][row * KPAD + c4 * 4];
            asm volatile("global_load_async_to_lds_b128 %0, %1, %2"
                         :: "v"(ldsK), "v"(goff), "s"(kg) : "memory");
            asm volatile("global_load_async_to_lds_b128 %0, %1, %2"
                         :: "v"(ldsV), "v"(goff), "s"(vg) : "memory");
        }
    };

    const int num_kt = qb / KTILE + 2;     // causal bound for this block
    stage_tile(0, 0);

    for (int kt = 0; kt < num_kt; ++kt) {
        const int buf = kt & 1;
        // Our DMAs for `buf` done; barrier makes the tile globally visible and
        // guarantees everyone finished reading buf^1 before we overwrite it.
        asm volatile("s_wait_asynccnt 0x0" ::: "memory");
        __syncthreads();
        if (kt + 1 < num_kt) stage_tile(kt + 1, buf ^ 1);   // overlap DMA with compute

        if (kt * KTILE > q15) continue;  // scalar branch: fully masked tile for this wave

        const float* KsB = Ks[buf];
        const float* VsB = Vs[buf];

        // ---- scores for the four 16-key subtiles (16 wmma each)
        v8f sacc4[4];
        #pragma unroll
        for (int st = 0; st < 4; ++st) {
            if (kt * KTILE + st * 16 <= q15) {              // scalar branch
                v8f s = (v8f)(0.0f);
                const float* kRow = &KsB[(st * 16 + l16) * KPAD];
                #pragma unroll
                for (int kb = 0; kb < 16; ++kb) {
                    const int k = 4 * kb + 2 * half;
                    v2f bf; bf.x = kRow[k]; bf.y = kRow[k + 1];
                    s = __builtin_amdgcn_wmma_f32_16x16x4_f32(
                        false, qf[kb], false, bf, (short)0, s, false, false);
                }
                sacc4[st] = s;
            } else {
                sacc4[st] = (v8f)(0.0f);   // causal select masks these rows to -inf anyway
            }
        }

        // ---- 64-key-wide causal mask + online softmax (one pass per K-tile)
        #pragma unroll
        for (int r = 0; r < 8; ++r) {
            const int qrow = q0 + r + 8 * half;
            float sv[4];
            #pragma unroll
            for (int st = 0; st < 4; ++st) {
                const int key = kt * KTILE + st * 16 + l16;
                sv[st] = (key <= qrow) ? sacc4[st][r] : -__builtin_inff();
            }
            const float msub = fmaxf(fmaxf(sv[0], sv[1]), fmaxf(sv[2], sv[3]));
            const float mx   = half16_max(msub);
            const float mnew = fmaxf(mrun[r], mx);
            const float alpha = exp2_(mrun[r] - mnew);      // first tile: exp2(-inf)=0
            mrun[r] = mnew;
            #pragma unroll
            for (int t = 0; t < 4; ++t) oacc[t][r] *= alpha;
            lacc[r] *= alpha;
            float* prow = pwave + (r + 8 * half) * PPAD;
            #pragma unroll
            for (int st = 0; st < 4; ++st)
                prow[st * 16 + l16] = exp2_(sv[st] - mnew);  // 0 for masked
        }

        // LDS ops from one wave are architecturally in-order; stop compiler reordering.
        asm volatile("" ::: "memory");

        // ---- out += P(16x64) * V(64x64), l += P * ones
        #pragma unroll
        for (int st = 0; st < 4; ++st) {
            if (kt * KTILE + st * 16 > q15) continue;       // scalar branch; P would be all-zero
            #pragma unroll
            for (int kb = 0; kb < 4; ++kb) {
                const int kcol = st * 16 + 4 * kb + 2 * half;
                v2f af;   // A-fragment of P: row=l16, K = kcol + {0,1}
                af.x = pwave[l16 * PPAD + kcol];
                af.y = pwave[l16 * PPAD + kcol + 1];
                #pragma unroll
                for (int t = 0; t < 4; ++t) {
                    v2f bf;   // B-fragment of V: N = d-within-tile = l16
                    bf.x = VsB[kcol * KPAD + t * 16 + l16];
                    bf.y = VsB[(kcol + 1) * KPAD + t * 16 + l16];
                    oacc[t] = __builtin_amdgcn_wmma_f32_16x16x4_f32(
                        false, af, false, bf, (short)0, oacc[t], false, false);
                }
                lacc = __builtin_amdgcn_wmma_f32_16x16x4_f32(
                    false, af, false, ones, (short)0, lacc, false, false);
            }
        }
    }

    // ---- normalize and store
    #pragma unroll
    for (int r = 0; r < 8; ++r) {
        const float inv = 1.0f / lacc[r];
        const int qrow = q0 + r + 8 * half;
        float* orow = O + base + (size_t)qrow * DHEAD;
        #pragma unroll
        for (int t = 0; t < 4; ++t)
            orow[t * 16 + l16] = oacc[t][r] * inv;
    }
}

extern "C" void kernel_launch(void* const* d_in, const int* in_sizes, int n_in,
                              void* d_out, int out_size, void* d_ws, size_t ws_size,
                              hipStream_t stream) {
    (void)in_sizes; (void)n_in; (void)out_size; (void)d_ws; (void)ws_size;
    const float* Q = (const float*)d_in[0];
    const float* K = (const float*)d_in[1];
    const float* V = (const float*)d_in[2];
    // d_in[3] is the causal mask; computed analytically in-kernel.
    float* O = (float*)d_out;

    dim3 grid(SEQ / QTILE_BLOCK, BHEADS);   // (16, 32)
    dim3 block(256);
    fa_fwd_f32_wmma<<<grid, block, 0, stream>>>(Q, K, V, O);
}